// GCN_37460704755812
// MI455X (gfx1250) — compile-verified
//
#include <hip/hip_runtime.h>
#include <hip/hip_bf16.h>

typedef float v2f __attribute__((ext_vector_type(2)));
typedef float v8f __attribute__((ext_vector_type(8)));

#define HID 128
#define HALFHID 64
#define LDS_STRIDE 132   // 128 + 4 pad: lane lid reads bank (4*lid + ka) % 64 -> conflict-free

// ---------------------------------------------------------------------------
// OUT[n,128] = relu((A [+ A2]) @ W[128,128] + b[128])
// grid.x = n/16 row strips, block = 256 threads = 8 waves; wave w owns cols
// [16w,16w+16). A strip staged in LDS (shared by all 8 waves, residual add
// fused); B strip preloaded to 64 VGPRs. Inner loop: ds_load_b64 + v_wmma.
// ---------------------------------------------------------------------------
template <bool RESID>
__global__ void gcn_gemm128_relu(const float* __restrict__ A,
                                 const float* __restrict__ A2,
                                 const float* __restrict__ W,
                                 const float* __restrict__ bias,
                                 float* __restrict__ OUT)
{
    __shared__ float As[16 * LDS_STRIDE];

    const int tid  = threadIdx.x;
    const int wave = tid >> 5;
    const int lane = tid & 31;
    const int half = lane >> 4;   // selects K pair within a 4-wide K step
    const int lid  = lane & 15;   // row (A) / col (B,C)
    const int row0 = blockIdx.x * 16;
    const int col0 = wave * 16;

    // ---- stage A(+A2) strip [16 x 128] into LDS: 512 float4s over 256 thr ----
    {
        const float4* Ag  = (const float4*)(A + (size_t)row0 * HID);
        const float4* Ag2 = RESID ? (const float4*)(A2 + (size_t)row0 * HID) : nullptr;
        #pragma unroll
        for (int i = 0; i < 2; ++i) {
            const int idx = tid + i * 256;        // [0,512)
            const int r   = idx >> 5;             // row 0..15
            const int c   = idx & 31;             // float4 col 0..31
            float4 v = Ag[idx];
            if (RESID) {
                const float4 u = Ag2[idx];
                v.x += u.x; v.y += u.y; v.z += u.z; v.w += u.w;
            }
            *(float4*)(As + r * LDS_STRIDE + c * 4) = v;
        }
    }

    // ---- preload B fragments for this wave's 16-col strip: 32 x v2f ----
    v2f bfrag[32];
    {
        const float* Wp = W + col0 + lid;
        #pragma unroll
        for (int k = 0; k < 32; ++k) {
            const int ka = 4 * k + 2 * half;
            bfrag[k].x = Wp[(size_t)(ka + 0) * HID];
            bfrag[k].y = Wp[(size_t)(ka + 1) * HID];
        }
    }

    __syncthreads();

    // ---- K loop: 32 x (ds_load_b64 + v_wmma_f32_16x16x4_f32) ----
    v8f acc = {};
    const float* arow = As + lid * LDS_STRIDE + 2 * half;
    #pragma unroll
    for (int k = 0; k < 32; ++k) {
        v2f afrag;
        afrag.x = arow[4 * k + 0];
        afrag.y = arow[4 * k + 1];
        acc = __builtin_amdgcn_wmma_f32_16x16x4_f32(
            false, afrag, false, bfrag[k], (short)0, acc, false, false);
    }

    // ---- epilogue: bias + relu + store ----
    const float bv = bias[col0 + lid];
    #pragma unroll
    for (int r = 0; r < 8; ++r) {
        const int row = row0 + r + 8 * half;
        float v = acc[r] + bv;
        OUT[(size_t)row * HID + col0 + lid] = v > 0.f ? v : 0.f;
    }
}

// ---------------------------------------------------------------------------
// HIDOUT[16*grid.x, 64] = relu(H[mask[p], :] @ W[128,64] + b[64])
// block = 128 threads = 4 waves (64 cols); rows gathered through post_mask.
// Tiny workload (512 blocks) -> direct global loads are fine.
// ---------------------------------------------------------------------------
__global__ void gcn_head1(const float* __restrict__ H,
                          const int* __restrict__ mask,
                          const float* __restrict__ W,
                          const float* __restrict__ bias,
                          float* __restrict__ HIDOUT)
{
    const int wave = threadIdx.x >> 5;
    const int lane = threadIdx.x & 31;
    const int half = lane >> 4;
    const int lid  = lane & 15;
    const int row0 = blockIdx.x * 16;
    const int col0 = wave * 16;

    const int node = mask[row0 + lid];
    const float* arow = H + (size_t)node * HID;
    const float* Wp   = W + col0 + lid;

    v8f acc = {};
    #pragma unroll 4
    for (int k = 0; k < 32; ++k) {
        const int ka = 4 * k + 2 * half;
        v2f afrag;
        afrag.x = arow[ka + 0];
        afrag.y = arow[ka + 1];
        v2f bfrag;
        bfrag.x = Wp[(size_t)(ka + 0) * HALFHID];
        bfrag.y = Wp[(size_t)(ka + 1) * HALFHID];
        acc = __builtin_amdgcn_wmma_f32_16x16x4_f32(
            false, afrag, false, bfrag, (short)0, acc, false, false);
    }

    const float bv = bias[col0 + lid];
    #pragma unroll
    for (int r = 0; r < 8; ++r) {
        const int row = row0 + r + 8 * half;
        float v = acc[r] + bv;
        HIDOUT[(size_t)row * HALFHID + col0 + lid] = v > 0.f ? v : 0.f;
    }
}

// ---------------------------------------------------------------------------
// logits -> sigmoid: out[p] = sigmoid(dot(hid[p,:64], w) + b)
// ---------------------------------------------------------------------------
__global__ void gcn_head2(const float* __restrict__ HIDIN,
                          const float* __restrict__ w,
                          const float* __restrict__ b,
                          float* __restrict__ out, int P)
{
    const int p = blockIdx.x * blockDim.x + threadIdx.x;
    if (p >= P) return;
    float s = b[0];
    const float* row = HIDIN + (size_t)p * HALFHID;
    #pragma unroll 8
    for (int i = 0; i < HALFHID; ++i) s += row[i] * w[i];
    out[p] = 1.0f / (1.0f + __expf(-s));
}

// ---------------------------------------------------------------------------
// AGG[dst[e], :] += H[src[e], :]   — one wave32 per edge, 4 floats per lane.
// Coalesced 512B row gather (mostly L2 hits: H is 25.6MB vs 192MB L2);
// scatter via global_atomic_add_f32 at the L2 atomic units.
// ---------------------------------------------------------------------------
__global__ void gcn_aggregate(const float* __restrict__ H,
                              const int* __restrict__ src,
                              const int* __restrict__ dst,
                              float* __restrict__ AGG, int n_edges)
{
    const int wid  = (int)((blockIdx.x * (size_t)blockDim.x + threadIdx.x) >> 5);
    const int lane = threadIdx.x & 31;
    if (wid >= n_edges) return;
    const int s = src[wid];
    const int d = dst[wid];
    const float4 v = ((const float4*)(H + (size_t)s * HID))[lane];
    float* ap = AGG + (size_t)d * HID + lane * 4;
    atomicAdd(ap + 0, v.x);
    atomicAdd(ap + 1, v.y);
    atomicAdd(ap + 2, v.z);
    atomicAdd(ap + 3, v.w);
}

__global__ void gcn_zero4(float4* __restrict__ p, long n4)
{
    const long i = blockIdx.x * (long)blockDim.x + threadIdx.x;
    if (i < n4) p[i] = make_float4(0.f, 0.f, 0.f, 0.f);
}

// ---------------------------------------------------------------------------
extern "C" void kernel_launch(void* const* d_in, const int* in_sizes, int n_in,
                              void* d_out, int out_size, void* d_ws, size_t ws_size,
                              hipStream_t stream)
{
    const float* X      = (const float*)d_in[0];   // [N,128]
    const int*   eidx   = (const int*)  d_in[1];   // [2,E]
    const int*   pmask  = (const int*)  d_in[2];   // [P]
    const float* W_enc  = (const float*)d_in[3];
    const float* b_enc  = (const float*)d_in[4];
    const float* W1     = (const float*)d_in[5];
    const float* b1     = (const float*)d_in[6];
    const float* W2     = (const float*)d_in[7];
    const float* b2     = (const float*)d_in[8];
    const float* W_o1   = (const float*)d_in[9];   // [128,64]
    const float* b_o1   = (const float*)d_in[10];
    const float* W_o2   = (const float*)d_in[11];  // [64]
    const float* b_o2   = (const float*)d_in[12];

    const int N = in_sizes[0] / HID;       // 50000 (multiple of 16)
    const int E = in_sizes[1] / 2;         // 1600000
    const int P = in_sizes[2];             // 8192  (multiple of 16)
    const int* src = eidx;
    const int* dst = eidx + E;

    const size_t node_elems = (size_t)N * HID;
    float* buf0 = (float*)d_ws;                 // h
    float* buf1 = buf0 + node_elems;            // agg (re-zeroed per pass)
    float* buf2 = buf1 + node_elems;            // h'
    float* hid  = buf2 + node_elems;            // [P,64]

    const int gemm_blocks = N / 16;             // 3125
    const long n4         = (long)(node_elems / 4);
    const int zero_blocks = (int)((n4 + 255) / 256);
    const int agg_blocks  = (E + 7) / 8;        // 8 wave-edges per 256-thr block
    float* out = (float*)d_out;

    // layer 1: h = relu(X @ W_enc + b_enc)
    gcn_gemm128_relu<false><<<gemm_blocks, 256, 0, stream>>>(X, nullptr, W_enc, b_enc, buf0);

    // agg = scatter_add(h)
    gcn_zero4<<<zero_blocks, 256, 0, stream>>>((float4*)buf1, n4);
    gcn_aggregate<<<agg_blocks, 256, 0, stream>>>(buf0, src, dst, buf1, E);

    // layer 2: h = relu((agg + h) @ W1 + b1)   (residual add fused into LDS staging)
    gcn_gemm128_relu<true><<<gemm_blocks, 256, 0, stream>>>(buf0, buf1, W1, b1, buf2);

    // agg = scatter_add(h)
    gcn_zero4<<<zero_blocks, 256, 0, stream>>>((float4*)buf1, n4);
    gcn_aggregate<<<agg_blocks, 256, 0, stream>>>(buf2, src, dst, buf1, E);

    // layer 3: h = relu((agg + h) @ W2 + b2)
    gcn_gemm128_relu<true><<<gemm_blocks, 256, 0, stream>>>(buf2, buf1, W2, b2, buf0);

    // head: gather posts, hid = relu(post @ W_o1 + b_o1)
    gcn_head1<<<P / 16, 128, 0, stream>>>(buf0, pmask, W_o1, b_o1, hid);

    // logits + sigmoid
    gcn_head2<<<(P + 255) / 256, 256, 0, stream>>>(hid, W_o2, b_o2, out, P);
}